// ALiBi2D_74904229642674
// MI455X (gfx1250) — compile-verified
//
#include <hip/hip_runtime.h>

// ALiBi-2D additive bias, MI455X (gfx1250, wave32).
// dist^2 tile via one V_WMMA_F32_16X16X4_F32 per 16x16 (i,j) tile (rank-4
// quadratic form); dist kept in 8 VGPRs and reused across all 16 heads.
// Streaming loop is explicitly software-pipelined (2-head batches, prefetch
// distance 1) so ~16 NT loads stay in flight per wave -> enough chip-wide
// outstanding bytes to sustain 23.3 TB/s. Roofline: ~46 us for 1 GB traffic.

typedef __attribute__((ext_vector_type(2))) float v2f;
typedef __attribute__((ext_vector_type(8))) float v8f;

#define T_DIM 2048
#define N_HEADS 16
#define TILES (T_DIM / 16)   // 128 tiles per axis

__global__ __launch_bounds__(64)
void alibi2d_wmma_kernel(const float* __restrict__ scores,
                         const int*   __restrict__ coords,   // [B,T,2]
                         const float* __restrict__ slopes,   // [16]
                         float*       __restrict__ out) {
  const int lane = threadIdx.x & 31;  // lane within wave32
  const int wave = threadIdx.x >> 5;  // 0 or 1: two waves per workgroup
  const int m    = lane & 15;         // row/col index within tile
  const int half = lane >> 4;         // 0: lanes 0-15, 1: lanes 16-31

  // blockIdx.x = ((b * TILES) + it) * (TILES/2) + jt_pair
  int bid = blockIdx.x;
  const int jt = ((bid & (TILES / 2 - 1)) << 1) | wave; bid >>= 6;
  const int it = bid & (TILES - 1);                     bid >>= 7;
  const int b  = bid;

  const int i0 = it * 16;
  const int j0 = jt * 16;

  // --- load coords for this tile's rows (i) and cols (j) -------------------
  const int* cb = coords + (size_t)b * T_DIM * 2;
  const int2 ci = *(const int2*)(cb + (size_t)(i0 + m) * 2);
  const int2 cj = *(const int2*)(cb + (size_t)(j0 + m) * 2);
  const float xi = (float)ci.x, yi = (float)ci.y;
  const float xj = (float)cj.x, yj = (float)cj.y;
  const float si = xi * xi + yi * yi;   // |c_i|^2  (exact: ints < 2^24)
  const float sj = xj * xj + yj * yj;   // |c_j|^2

  // --- A (16x4), B (4x16) for the quadratic form ---------------------------
  // A row M = ( xi, yi, si, 1 ),  B col N = ( -2xj, -2yj, 1, sj )
  // => (A x B)[i][j] = si + sj - 2 xi xj - 2 yi yj = dist^2(i,j), exact.
  // f32 A 16x4 layout: lanes 0-15 -> M, VGPR0=K0, VGPR1=K1; lanes 16-31 K2,K3.
  v2f av, bv;
  if (half == 0) {
    av.x = xi;          av.y = yi;
    bv.x = -2.0f * xj;  bv.y = -2.0f * yj;
  } else {
    av.x = si;          av.y = 1.0f;
    bv.x = 1.0f;        bv.y = sj;
  }

  // --- one WMMA: D(16x16) = dist^2 tile ------------------------------------
  v8f acc = {};
  acc = __builtin_amdgcn_wmma_f32_16x16x4_f32(
      /*neg_a=*/false, av, /*neg_b=*/false, bv,
      /*c_mod=*/(short)0, acc, /*reuse_a=*/false, /*reuse_b=*/false);

  // dist = sqrt(dist^2): single v_sqrt_f32 (exact small-integer input).
  // Accumulator element r is (M = r + 8*half, N = m).
  float dist[8];
#pragma unroll
  for (int r = 0; r < 8; ++r) {
    dist[r] = __builtin_amdgcn_sqrtf(acc[r]);
  }

  // --- stream all 16 heads; 2-head batches, prefetch next batch ------------
  const size_t head_stride = (size_t)T_DIM * T_DIM;
  const size_t row_stride  = T_DIM;
  const size_t base =
      (((size_t)b * N_HEADS) * T_DIM + (size_t)(i0 + half * 8)) * T_DIM
      + (size_t)(j0 + m);

  const float* __restrict__ sp = scores + base;
  float*       __restrict__ op = out + base;

  float cur[16], nxt[16];

  // prime the pipeline: heads 0 and 1 (16 loads issued back-to-back)
#pragma unroll
  for (int r = 0; r < 8; ++r) {
    cur[r]     = __builtin_nontemporal_load(sp + (size_t)r * row_stride);
    cur[8 + r] = __builtin_nontemporal_load(sp + head_stride
                                            + (size_t)r * row_stride);
  }

#pragma unroll
  for (int hh = 0; hh < N_HEADS; hh += 2) {
    // prefetch heads hh+2, hh+3 while current batch's stores drain
    if (hh + 2 < N_HEADS) {
#pragma unroll
      for (int r = 0; r < 8; ++r) {
        nxt[r]     = __builtin_nontemporal_load(
            sp + (size_t)(hh + 2) * head_stride + (size_t)r * row_stride);
        nxt[8 + r] = __builtin_nontemporal_load(
            sp + (size_t)(hh + 3) * head_stride + (size_t)r * row_stride);
      }
    }

    const float ns0 = -slopes[hh];       // uniform -> scalar loads
    const float ns1 = -slopes[hh + 1];
#pragma unroll
    for (int r = 0; r < 8; ++r) {
      __builtin_nontemporal_store(
          __builtin_fmaf(ns0, dist[r], cur[r]),
          op + (size_t)hh * head_stride + (size_t)r * row_stride);
      __builtin_nontemporal_store(
          __builtin_fmaf(ns1, dist[r], cur[8 + r]),
          op + (size_t)(hh + 1) * head_stride + (size_t)r * row_stride);
    }

#pragma unroll
    for (int r = 0; r < 16; ++r) cur[r] = nxt[r];  // SSA-renamed away
  }
}

extern "C" void kernel_launch(void* const* d_in, const int* in_sizes, int n_in,
                              void* d_out, int out_size, void* d_ws, size_t ws_size,
                              hipStream_t stream) {
  const float* scores = (const float*)d_in[0];
  const int*   coords = (const int*)d_in[1];
  const float* slopes = (const float*)d_in[2];
  float*       out    = (float*)d_out;

  const int B = in_sizes[1] / (T_DIM * 2);          // coords_xy is [B,T,2]
  const int nblocks = B * TILES * (TILES / 2);      // 2 waves (j-tiles) / block

  alibi2d_wmma_kernel<<<nblocks, 64, 0, stream>>>(scores, coords, slopes, out);
}